// HHGNN_hetero_9371618640200
// MI455X (gfx1250) — compile-verified
//
#include <hip/hip_runtime.h>
#include <hip/hip_bf16.h>

// ---------------------------------------------------------------------------
// HHGNN hetero pipeline for gfx1250 (MI455X), wave32 + V_WMMA_F32_16X16X4_F32
// ---------------------------------------------------------------------------

#define NNODES 12288
#define SSEG   4096
#define MEDGES 16384
#define EINC   393216
#define FCH    256
#define CCH    128
#define SLOPE  0.2f

typedef float v2f __attribute__((ext_vector_type(2)));
typedef float v8f __attribute__((ext_vector_type(8)));

__device__ __forceinline__ float lrelu(float v) {
    return v >= 0.0f ? v : SLOPE * v;
}

__device__ __forceinline__ void atomicMaxF(float* addr, float val) {
    int* ai = (int*)addr;
    int cur = __float_as_int(*addr);
    while (__int_as_float(cur) < val) {
        int old = atomicCAS(ai, cur, __float_as_int(val));
        if (old == cur) break;
        cur = old;
    }
}

// ---------------------------------------------------------------------------
// fp32 WMMA GEMM, templated on B layout and K (compile-time trip count):
//   C[M x N] = A[M x K] @ B (+ bias)(+ LeakyReLU)
//   BT == 0 : B row-major K x N, element (k,n) = B[k*ldb + n]
//   BT == 1 : B row-major N x K (A @ B^T), element (k,n) = B[n*ldb + k]
// Each wave computes a 16(M) x 64(N) strip: one A fragment feeds 4 WMMAs.
// Block = (32,4) waves -> 64x64 C tile per workgroup. Grid = (N/64, M/64).
// fp32 fragment layout (ISA 7.12.2):
//   A 16x4 : lane L (r=L&15, hi=L>>4) holds A[r][k+2*hi], A[r][k+2*hi+1]
//   B 4x16 : lane L holds B[k+2*hi][c=r], B[k+2*hi+1][c=r]
//   C 16x16: vgpr i -> row = i + 8*hi, col = r
// ---------------------------------------------------------------------------
template <int BT, int K>
__global__ void k_gemm_wmma(const float* __restrict__ A,
                            const float* __restrict__ B,
                            const float* __restrict__ bias,
                            float* __restrict__ C,
                            int ldb, int ldc, int act) {
    const int lane    = threadIdx.x;      // 0..31
    const int r       = lane & 15;
    const int hi      = lane >> 4;
    const int rowBase = (blockIdx.y * 4 + threadIdx.y) * 16;
    const int colBase = blockIdx.x * 64;

    const float* Arow = A + (size_t)(rowBase + r) * K;
    const float* Bp0, *Bp1, *Bp2, *Bp3;
    if (BT) {   // rows of B^T (contiguous in k)
        Bp0 = B + (size_t)(colBase + r) * ldb;
        Bp1 = B + (size_t)(colBase + 16 + r) * ldb;
        Bp2 = B + (size_t)(colBase + 32 + r) * ldb;
        Bp3 = B + (size_t)(colBase + 48 + r) * ldb;
    } else {    // columns of B (stride ldb in k)
        Bp0 = B + (colBase + r);
        Bp1 = B + (colBase + 16 + r);
        Bp2 = B + (colBase + 32 + r);
        Bp3 = B + (colBase + 48 + r);
    }

    v8f acc0 = {}, acc1 = {}, acc2 = {}, acc3 = {};
#pragma unroll 4
    for (int k = 0; k < K; k += 4) {
        const int kk = k + 2 * hi;
        v2f a, b0, b1, b2, b3;
        a.x = Arow[kk];
        a.y = Arow[kk + 1];
        if (BT) {
            b0.x = Bp0[kk]; b0.y = Bp0[kk + 1];
            b1.x = Bp1[kk]; b1.y = Bp1[kk + 1];
            b2.x = Bp2[kk]; b2.y = Bp2[kk + 1];
            b3.x = Bp3[kk]; b3.y = Bp3[kk + 1];
        } else {
            b0.x = Bp0[(size_t)kk * ldb]; b0.y = Bp0[(size_t)(kk + 1) * ldb];
            b1.x = Bp1[(size_t)kk * ldb]; b1.y = Bp1[(size_t)(kk + 1) * ldb];
            b2.x = Bp2[(size_t)kk * ldb]; b2.y = Bp2[(size_t)(kk + 1) * ldb];
            b3.x = Bp3[(size_t)kk * ldb]; b3.y = Bp3[(size_t)(kk + 1) * ldb];
        }
        __builtin_prefetch(Arow + k + 32, 0, 3);   // global_prefetch_b8, near cache
        acc0 = __builtin_amdgcn_wmma_f32_16x16x4_f32(false, a, false, b0, (short)0, acc0, false, false);
        acc1 = __builtin_amdgcn_wmma_f32_16x16x4_f32(false, a, false, b1, (short)0, acc1, false, false);
        acc2 = __builtin_amdgcn_wmma_f32_16x16x4_f32(false, a, false, b2, (short)0, acc2, false, false);
        acc3 = __builtin_amdgcn_wmma_f32_16x16x4_f32(false, a, false, b3, (short)0, acc3, false, false);
    }

#pragma unroll
    for (int j = 0; j < 4; ++j) {
        const v8f& acc = j == 0 ? acc0 : j == 1 ? acc1 : j == 2 ? acc2 : acc3;
        const int col = colBase + 16 * j + r;
        const float bv = bias ? bias[col] : 0.0f;
#pragma unroll
        for (int i = 0; i < 8; ++i) {
            const int row = rowBase + i + 8 * hi;
            float v = acc[i] + bv;
            if (act) v = lrelu(v);
            C[(size_t)row * ldc + col] = v;
        }
    }
}

// out[row] = dot(X[row, 0:K], att[0:K]); one wave per row, 8 floats/lane
__global__ void k_rowdot(const float* __restrict__ X,
                         const float* __restrict__ att,
                         float* __restrict__ out, int rows, int K) {
    const int lane = threadIdx.x;
    const int row  = blockIdx.x * blockDim.y + threadIdx.y;
    if (row >= rows) return;
    const float* xr = X + (size_t)row * K;
    float s = 0.0f;
    for (int c0 = lane * 8; c0 < K; c0 += 32 * 8) {
        float4 xv0 = *(const float4*)(xr + c0);
        float4 xv1 = *(const float4*)(xr + c0 + 4);
        float4 av0 = *(const float4*)(att + c0);
        float4 av1 = *(const float4*)(att + c0 + 4);
        s += xv0.x * av0.x + xv0.y * av0.y + xv0.z * av0.z + xv0.w * av0.w;
        s += xv1.x * av1.x + xv1.y * av1.y + xv1.z * av1.z + xv1.w * av1.w;
    }
#pragma unroll
    for (int off = 16; off > 0; off >>= 1) s += __shfl_xor(s, off, 32);
    if (lane == 0) out[row] = s;
}

__global__ void k_fill(float* __restrict__ p, float v, int n) {
    int i = blockIdx.x * blockDim.x + threadIdx.x;
    if (i < n) p[i] = v;
}

// out[row][c] = b0[c] + b1[c] + b2[c] for every row (bias pre-accumulation)
__global__ void k_bias3(float* __restrict__ out,
                        const float* __restrict__ b0,
                        const float* __restrict__ b1,
                        const float* __restrict__ b2, int total, int F) {
    int i = blockIdx.x * blockDim.x + threadIdx.x;
    if (i < total) {
        int c = i & (F - 1);
        out[i] = b0[c] + b1[c] + b2[c];
    }
}

__global__ void k_leaky(float* __restrict__ p, int n) {
    int i = blockIdx.x * blockDim.x + threadIdx.x;
    if (i < n) p[i] = lrelu(p[i]);
}

__global__ void k_copy(float* __restrict__ dst, const float* __restrict__ src, int n) {
    int i = blockIdx.x * blockDim.x + threadIdx.x;
    if (i < n) dst[i] = src[i];
}

// pass 1: attention logit + segment max over node groups
__global__ void k_edge1(const int* __restrict__ ni, const int* __restrict__ ei,
                        const float* __restrict__ sx, const float* __restrict__ se,
                        float* __restrict__ amax, int E) {
    int e = blockIdx.x * blockDim.x + threadIdx.x;
    if (e >= E) return;
    const int n = ni[e];
    const float a = lrelu(sx[n] + se[ei[e]]);
    atomicMaxF(&amax[n], a);
}

// pass 2: exp(a - amax), softmax denominator, degree sums
__global__ void k_edge2(const int* __restrict__ ni, const int* __restrict__ ei,
                        const float* __restrict__ sx, const float* __restrict__ se,
                        const float* __restrict__ amax, const float* __restrict__ hw,
                        float* __restrict__ ae, float* __restrict__ asum,
                        float* __restrict__ Dn, float* __restrict__ Bc, int E) {
    int e = blockIdx.x * blockDim.x + threadIdx.x;
    if (e >= E) return;
    const int n = ni[e];
    const int m = ei[e];
    const float a = lrelu(sx[n] + se[m]);
    float am = amax[n];
    if (!isfinite(am)) am = 0.0f;          // reference: where(isfinite, amax, 0)
    const float v = expf(a - am);
    ae[e] = v;
    atomicAdd(&asum[n], v);
    atomicAdd(&Dn[n], hw[m]);
    atomicAdd(&Bc[m], 1.0f);
}

// weighted 256-wide row scatter-add: dst[m] += w * src[n]; 8 floats per lane
__device__ __forceinline__ void rowScatter(const float* __restrict__ src,
                                           float* __restrict__ dst,
                                           float w, int lane) {
    const int c = lane * 8;
    float4 s0 = *(const float4*)(src + c);
    float4 s1 = *(const float4*)(src + c + 4);
    float* d = dst + c;
    atomicAdd(d + 0, w * s0.x);
    atomicAdd(d + 1, w * s0.y);
    atomicAdd(d + 2, w * s0.z);
    atomicAdd(d + 3, w * s0.w);
    atomicAdd(d + 4, w * s1.x);
    atomicAdd(d + 5, w * s1.y);
    atomicAdd(d + 6, w * s1.z);
    atomicAdd(d + 7, w * s1.w);
}

// pass 3: medge[m] += (Binv * alpha) * xl[n]   (one wave per incidence)
__global__ void k_scat_edges(const int* __restrict__ ni, const int* __restrict__ ei,
                             const float* __restrict__ ae, const float* __restrict__ asum,
                             const float* __restrict__ Bc, const float* __restrict__ xl,
                             float* __restrict__ medge, int E) {
    const int lane = threadIdx.x;
    const int e = blockIdx.x * blockDim.y + threadIdx.y;
    if (e >= E) return;
    const int n = ni[e];
    const int m = ei[e];
    const float as = asum[n];
    const float alpha = as > 0.0f ? ae[e] / as : 0.0f;
    const float bc = Bc[m];
    const float w = bc > 0.0f ? alpha / bc : 0.0f;
    rowScatter(xl + (size_t)n * FCH, medge + (size_t)m * FCH, w, lane);
}

// pass 4: hout[n] += (Dinv * alpha) * medge[m]
__global__ void k_scat_nodes(const int* __restrict__ ni, const int* __restrict__ ei,
                             const float* __restrict__ ae, const float* __restrict__ asum,
                             const float* __restrict__ Dn, const float* __restrict__ medge,
                             float* __restrict__ hout, int E) {
    const int lane = threadIdx.x;
    const int e = blockIdx.x * blockDim.y + threadIdx.y;
    if (e >= E) return;
    const int n = ni[e];
    const int m = ei[e];
    const float as = asum[n];
    const float alpha = as > 0.0f ? ae[e] / as : 0.0f;
    const float d = Dn[n];
    const float w = d > 0.0f ? alpha / d : 0.0f;
    rowScatter(medge + (size_t)m * FCH, hout + (size_t)n * FCH, w, lane);
}

// ---------------------------------------------------------------------------

extern "C" void kernel_launch(void* const* d_in, const int* in_sizes, int n_in,
                              void* d_out, int out_size, void* d_ws, size_t ws_size,
                              hipStream_t stream) {
    const float* g  = (const float*)d_in[0];
    const float* x  = (const float*)d_in[1];
    const float* hw = (const float*)d_in[2];
    const float* hyperAttr = (const float*)d_in[3];
    const int* ni[3] = {(const int*)d_in[4], (const int*)d_in[6], (const int*)d_in[8]};
    const int* ei[3] = {(const int*)d_in[5], (const int*)d_in[7], (const int*)d_in[9]};
    const float* W0   = (const float*)d_in[10];
    const float* b0   = (const float*)d_in[11];
    const float* Wh1  = (const float*)d_in[12];
    const float* att1 = (const float*)d_in[13];
    const float* bc1  = (const float*)d_in[14];
    const float* W1   = (const float*)d_in[15];
    const float* b1   = (const float*)d_in[16];
    const float* Wh2  = (const float*)d_in[17];
    const float* att2 = (const float*)d_in[18];
    const float* bc2  = (const float*)d_in[19];
    const float* Wg   = (const float*)d_in[20];
    const float* bg   = (const float*)d_in[21];
    const float* Wx   = (const float*)d_in[22];
    const float* bx   = (const float*)d_in[23];
    float* out = (float*)d_out;
    (void)in_sizes; (void)n_in; (void)out_size; (void)ws_size;

    // ---- workspace layout (floats) ----
    float* ws = (float*)d_ws;
    size_t off = 0;
    auto alloc = [&](size_t n) { float* p = ws + off; off += n; return p; };
    float* hA   = alloc((size_t)NNODES * FCH);
    float* hB   = alloc((size_t)NNODES * FCH);
    float* xl   = alloc((size_t)NNODES * FCH);
    float* ea   = alloc((size_t)MEDGES * FCH);
    float* me   = alloc((size_t)MEDGES * FCH);
    float* sx   = alloc(NNODES);
    float* se   = alloc(MEDGES);
    float* amax = alloc(NNODES);
    float* asum = alloc(NNODES);
    float* Dn   = alloc(NNODES);
    float* Bc   = alloc(MEDGES);
    float* ae   = alloc(EINC);
    float* ng1  = alloc((size_t)SSEG * CCH);
    float* ng2  = alloc((size_t)SSEG * CCH);
    float* xs1  = alloc((size_t)SSEG * CCH);
    float* xs2  = alloc((size_t)SSEG * CCH);

    const float NEG_INF = -__builtin_inff();

    // NN GEMM, K=256: C[M x N] = A @ B(row-major 256 x N) + bias
    auto gemmNN = [&](const float* A, const float* B, const float* bias, float* C,
                      int M, int N, int ldc, int act) {
        dim3 blk(32, 4);
        dim3 grd(N / 64, M / 64);
        k_gemm_wmma<0, 256><<<grd, blk, 0, stream>>>(A, B, bias, C, N, ldc, act);
    };
    // NT GEMM, K=128: C[M x N] = A @ B^T, B row-major N x 128
    auto gemmNT = [&](const float* A, const float* B, float* C,
                      int M, int N, int ldc) {
        dim3 blk(32, 4);
        dim3 grd(N / 64, M / 64);
        k_gemm_wmma<1, 128><<<grd, blk, 0, stream>>>(A, B, nullptr, C, CCH, ldc, 0);
    };
    auto fill = [&](float* p, float v, int n) {
        k_fill<<<(n + 255) / 256, 256, 0, stream>>>(p, v, n);
    };

    // per-segment Linear + LeakyReLU (3 independent 4096x256x256 GEMMs)
    auto seglin = [&](const float* in, const float* W, const float* b, float* outp) {
        for (int i = 0; i < 3; ++i)
            gemmNN(in + (size_t)i * SSEG * FCH, W + (size_t)i * FCH * FCH, b + i * FCH,
                   outp + (size_t)i * SSEG * FCH, SSEG, FCH, FCH, 1);
    };

    // sum of 3 HypergraphConv branches + LeakyReLU
    auto hconv3 = [&](const float* hin, float* hout, const float* Wh,
                      const float* att, const float* bc) {
        const int tot = NNODES * FCH;
        k_bias3<<<(tot + 255) / 256, 256, 0, stream>>>(hout, bc, bc + FCH, bc + 2 * FCH,
                                                       tot, FCH);
        for (int k = 0; k < 3; ++k) {
            const float* Wk   = Wh + (size_t)k * FCH * FCH;
            const float* attk = att + (size_t)k * 2 * FCH;
            gemmNN(hin, Wk, nullptr, xl, NNODES, FCH, FCH, 0);
            gemmNN(hyperAttr, Wk, nullptr, ea, MEDGES, FCH, FCH, 0);
            k_rowdot<<<NNODES / 8, dim3(32, 8), 0, stream>>>(xl, attk, sx, NNODES, FCH);
            k_rowdot<<<MEDGES / 8, dim3(32, 8), 0, stream>>>(ea, attk + FCH, se, MEDGES, FCH);
            fill(amax, NEG_INF, NNODES);
            fill(asum, 0.0f, NNODES);
            fill(Dn, 0.0f, NNODES);
            fill(Bc, 0.0f, MEDGES);
            fill(me, 0.0f, MEDGES * FCH);
            k_edge1<<<EINC / 256, 256, 0, stream>>>(ni[k], ei[k], sx, se, amax, EINC);
            k_edge2<<<EINC / 256, 256, 0, stream>>>(ni[k], ei[k], sx, se, amax, hw,
                                                    ae, asum, Dn, Bc, EINC);
            k_scat_edges<<<EINC / 8, dim3(32, 8), 0, stream>>>(ni[k], ei[k], ae, asum,
                                                               Bc, xl, me, EINC);
            k_scat_nodes<<<EINC / 8, dim3(32, 8), 0, stream>>>(ni[k], ei[k], ae, asum,
                                                               Dn, me, hout, EINC);
        }
        k_leaky<<<(tot + 255) / 256, 256, 0, stream>>>(hout, tot);
    };

    // ---- pipeline ----
    seglin(g, W0, b0, hA);                 // layer0
    hconv3(hA, hB, Wh1, att1, bc1);        // hyper1
    seglin(hB, W1, b1, hA);                // layer1
    hconv3(hA, hB, Wh2, att2, bc2);        // hyper2  -> final h in hB

    // projections to common space (only segments 1 and 2 feed the result)
    gemmNN(hB + (size_t)1 * SSEG * FCH, Wg + (size_t)1 * FCH * CCH, bg + 1 * CCH, ng1,
           SSEG, CCH, CCH, 1);
    gemmNN(hB + (size_t)2 * SSEG * FCH, Wg + (size_t)2 * FCH * CCH, bg + 2 * CCH, ng2,
           SSEG, CCH, CCH, 1);
    gemmNN(x, Wx + (size_t)1 * FCH * CCH, bx + 1 * CCH, xs1, SSEG, CCH, CCH, 1);
    gemmNN(x, Wx + (size_t)2 * FCH * CCH, bx + 2 * CCH, xs2, SSEG, CCH, CCH, 1);

    // result = [xs1 @ ng1^T | xs2 @ ng2^T]  -> [4096, 8192]
    gemmNT(xs1, ng1, out,        SSEG, SSEG, 2 * SSEG);
    gemmNT(xs2, ng2, out + SSEG, SSEG, SSEG, 2 * SSEG);

    // append h
    const int hn = NNODES * FCH;
    k_copy<<<(hn + 255) / 256, 256, 0, stream>>>(out + (size_t)SSEG * 2 * SSEG, hB, hn);
}